// FilteredNoise_66967130079917
// MI455X (gfx1250) — compile-verified
//
#include <hip/hip_runtime.h>
#include <hip/hip_bf16.h>

// ---------------------------------------------------------------------------
// FilteredNoise on gfx1250:
//   ir = Mmat(129x65) @ H  (exact irfft+roll+window as a fixed matmul) -> WMMA f32
//   filtered = conv(noise*2-1, ir) * 0.01   (L=192 == linear conv, no FFT)
//   overlap-add hop 64 via native fp32 atomics (agent scope, relaxed)
// ---------------------------------------------------------------------------

typedef __attribute__((ext_vector_type(2))) float v2f;
typedef __attribute__((ext_vector_type(8))) float v8f;

#define BATCH     32
#define FRAMES    4000
#define CBINS     65     // rfft bins
#define NIR       129    // IR length (2*C-1)
#define FRAME_LEN 64
#define SEGLEN    192    // FRAME_LEN + NIR - 1
#define OUTLEN    256128 // (FRAMES-1)*64 + 192
#define KPAD      68     // CBINS padded to multiple of 4
#define JROWS     144    // NIR padded to multiple of 16 (9 tiles)
#define IRS       145    // LDS row stride for IR (odd -> spread banks)
#define FPB       16     // frames per workgroup

// ---- build the fixed irfft/roll/window matrix, zero padded to 144x68 -------
__global__ void FN_setup_M(float* __restrict__ M) {
  const float TWO_PI = 6.28318530717958647692f;
  for (int i = blockIdx.x * blockDim.x + threadIdx.x; i < JROWS * KPAD;
       i += gridDim.x * blockDim.x) {
    int j = i / KPAD;
    int k = i - j * KPAD;
    float v = 0.0f;
    if (j < NIR && k < CBINS) {
      float win = 0.5f * (1.0f - cosf(TWO_PI * (float)j / (float)NIR));
      int m = (k * (j - 64)) % NIR;              // exact angle reduction
      float ang = TWO_PI * (float)m / (float)NIR;
      float ck  = (k == 0) ? 1.0f : 2.0f;
      v = win * (ck / (float)NIR) * cosf(ang);
    }
    M[i] = v;
  }
}

// ---- zero the output (harness poisons it with 0xAA); b128 stores -----------
__global__ void FN_zero(float4* __restrict__ out4, int n4) {
  float4 z; z.x = z.y = z.z = z.w = 0.0f;
  for (int i = blockIdx.x * blockDim.x + threadIdx.x; i < n4;
       i += gridDim.x * blockDim.x)
    out4[i] = z;
}

// ---- fused: WMMA ir-matmul into LDS, then conv + overlap-add ---------------
__global__ __launch_bounds__(256) void FN_fused(const float* __restrict__ H,
                                                const float* __restrict__ noise,
                                                const float* __restrict__ M,
                                                float* __restrict__ out) {
  __shared__ float irs[FPB * IRS];      // ir[frame][j], j in [0,144)
  __shared__ float nz[FPB * FRAME_LEN]; // noise mapped to [-1,1)

  const int tid  = threadIdx.x;
  const int lane = tid & 31;
  const int wave = tid >> 5;

  const int b  = blockIdx.x / (FRAMES / FPB);
  const int f0 = (blockIdx.x - b * (FRAMES / FPB)) * FPB;
  const long r0 = (long)b * FRAMES + f0;       // global frame row base

  // stage 0: stage noise into LDS, mapped to [-1,1)
  for (int i = tid; i < FPB * FRAME_LEN; i += 256) {
    int lf = i >> 6;
    int tt = i & 63;
    nz[i] = noise[(r0 + lf) * FRAME_LEN + tt] * 2.0f - 1.0f;
  }

  // stage 1: IR(16x144) = H(16x65) x Mmat^T via V_WMMA_F32_16X16X4_F32
  // A 16x4 f32 layout: row = lane&15, K pair = 2*(lane>>4)   (ISA 7.12.2)
  // B 4x16 f32 layout: col = lane&15, K pair = 2*(lane>>4)
  const int row = lane & 15;
  const int kb  = (lane >> 4) << 1;
  const float* hrow = H + (r0 + row) * CBINS;

  for (int jt = wave; jt < 9; jt += 8) {       // wave-uniform: EXEC stays full
    v8f acc = {};
    const float* mrow = M + (jt * 16 + row) * KPAD + kb;
#pragma unroll
    for (int k0 = 0; k0 < KPAD; k0 += 4) {
      int ka = k0 + kb;
      v2f a;
      a.x = (ka     < CBINS) ? hrow[ka]     : 0.0f;
      a.y = (ka + 1 < CBINS) ? hrow[ka + 1] : 0.0f;
      v2f bv = *(const v2f*)(mrow + k0);       // padded rows: no guard needed
      acc = __builtin_amdgcn_wmma_f32_16x16x4_f32(
          /*neg_a=*/false, a, /*neg_b=*/false, bv,
          /*c_mod=*/(short)0, acc, /*reuse_a=*/false, /*reuse_b=*/false);
    }
    // D 16x16 f32: M = v + 8*(lane>>4), N = lane&15
    const int mbase = (lane >> 4) << 3;
#pragma unroll
    for (int v = 0; v < 8; ++v)
      irs[(mbase + v) * IRS + jt * 16 + row] = acc[v];
  }

  __syncthreads();

  // stage 2: per-frame linear convolution + overlap-add (hop = 64)
  for (int o = tid; o < FPB * SEGLEN; o += 256) {
    int lf = o / SEGLEN;
    int t  = o - lf * SEGLEN;
    int lo = (t > NIR - 1) ? (t - (NIR - 1)) : 0;   // tau range so 0<=t-tau<129
    int hi = (t < FRAME_LEN - 1) ? t : (FRAME_LEN - 1);
    const float* np  = nz + (lf << 6);
    const float* irp = irs + lf * IRS + t;          // irp[-tau] = ir[t-tau]
    float s = 0.0f;
    for (int tau = lo; tau <= hi; ++tau)
      s += np[tau] * irp[-tau];
    // native fp32 atomic add (agent scope, relaxed) -> global_atomic_add_f32
    __hip_atomic_fetch_add(
        out + (long)b * OUTLEN + (long)(f0 + lf) * FRAME_LEN + t,
        s * 0.01f, __ATOMIC_RELAXED, __HIP_MEMORY_SCOPE_AGENT);
  }
}

extern "C" void kernel_launch(void* const* d_in, const int* in_sizes, int n_in,
                              void* d_out, int out_size, void* d_ws, size_t ws_size,
                              hipStream_t stream) {
  const float* H     = (const float*)d_in[0];  // (32,4000,65) f32
  const float* noise = (const float*)d_in[1];  // (32,4000,64) f32
  float* out = (float*)d_out;                  // (32,256128)  f32
  float* Mws = (float*)d_ws;                   // 144*68 floats scratch

  FN_setup_M<<<(JROWS * KPAD + 255) / 256, 256, 0, stream>>>(Mws);
  FN_zero<<<4096, 256, 0, stream>>>((float4*)out, out_size / 4);
  FN_fused<<<BATCH * (FRAMES / FPB), 256, 0, stream>>>(H, noise, Mws, out);
}